// SDGCN_with_ln_res_69295002353944
// MI455X (gfx1250) — compile-verified
//
#include <hip/hip_runtime.h>
#include <hip/hip_bf16.h>

typedef __attribute__((ext_vector_type(2))) float v2f;
typedef __attribute__((ext_vector_type(4))) float v4f;
typedef __attribute__((ext_vector_type(8))) float v8f;

#define NN 1024
#define DD 64
#define LDSTRIDE 68   // padded row stride (floats) for 64-wide rows: bank = (4r+c)%64, conflict-free
#define TSTRIDE  18   // padded row stride (floats) for 16-wide tiles: bank = (18r+c)%64, conflict-free

// LDS float layout:
//   [0     .. 1087]  : Xc tile buffer 0   16 x LDSTRIDE
//   [1088  .. 2175]  : Xc tile buffer 1   16 x LDSTRIDE
//   [2176  .. 6527]  : W                  64 x LDSTRIDE
//   [6528  .. +8*1088): per-wave scratch, 1088 floats each
//      m-loop :  PA tile 16xTSTRIDE at +0, A tile 16xTSTRIDE at +320
//      epilog :  H staging 16 x LDSTRIDE
#define LDS_FLOATS (2176 + 4352 + 8*1088)

#define AS3 __attribute__((address_space(3)))

// Async global->LDS copy, 16B per lane (GLOBAL_LOAD_ASYNC_TO_LDS_B128, ASYNCcnt).
__device__ __forceinline__ void async_copy_b128(float* lds_dst, const float* gsrc) {
    unsigned lds_off = (unsigned)(unsigned long long)(AS3 float*)lds_dst;
    unsigned long long ga = (unsigned long long)gsrc;
    asm volatile("global_load_async_to_lds_b128 %0, %1, off"
                 :: "v"(lds_off), "v"(ga) : "memory");
}
__device__ __forceinline__ void wait_async_le1() {
    asm volatile("s_wait_asynccnt 0x1" ::: "memory");
}
__device__ __forceinline__ void wait_async_le0() {
    asm volatile("s_wait_asynccnt 0x0" ::: "memory");
}

__global__ __launch_bounds__(256)
void sdgcn_fused_kernel(const float* __restrict__ X, const float* __restrict__ A,
                        const float* __restrict__ W, const float* __restrict__ gamma,
                        const float* __restrict__ beta, float* __restrict__ out)
{
    __shared__ float lds[LDS_FLOATS];
    float* wl = lds + 2176;                // 64 x 68

    const int tid  = threadIdx.x;
    const int wave = tid >> 5;
    const int lane = tid & 31;
    const int ln16 = lane & 15;
    const int hi   = lane >> 4;            // lane half: 0 or 1
    float* wsc = lds + 6528 + wave * 1088; // per-wave scratch

    const int slice   = blockIdx.x >> 3;   // 0..47  (b*T+t)
    const int rblk    = blockIdx.x & 7;    // 0..7
    const int rowbase = rblk * 128 + wave * 16;

    const float* Xs = X   + (size_t)slice * NN * DD;
    float*       Os = out + (size_t)slice * NN * DD;

    // Per-thread slice of the cooperative tile copy: 16B each, 256 threads = 4KB tile
    const int tr = tid >> 4;               // tile row 0..15
    const int tc = (tid & 15) * 4;         // tile col 0,4,..,60

    // ---- kick off async prefetch of tile 0 into buffer 0 ----
    async_copy_b128(lds + tr * LDSTRIDE + tc, Xs + (size_t)tr * DD + tc);

    // ---- preload W into LDS (row-major, padded stride), 16 floats/thread ----
    #pragma unroll
    for (int i = 0; i < 4; ++i) {
        int flat = tid * 4 + i * 1024;     // 0..4095
        int r = flat >> 6, c = flat & 63;
        v4f w4 = *(const v4f*)(W + flat);
        *(v4f*)(wl + r * LDSTRIDE + c) = w4;
    }

    // ---- preload my 16 rows of X as WMMA A-operand fragments ----
    // A-operand (16x4 fp32, K-chunk kk): lane holds Xrow[ln16][4kk + 2hi + {0,1}]
    v2f ax[16];
    {
        const float* xr = Xs + (size_t)(rowbase + ln16) * DD + 2 * hi;
        #pragma unroll
        for (int kk = 0; kk < 16; ++kk) ax[kk] = *(const v2f*)(xr + 4 * kk);
    }

    float gmm[4], bet[4];
    #pragma unroll
    for (int c = 0; c < 4; ++c) { gmm[c] = gamma[16*c + ln16]; bet[c] = beta[16*c + ln16]; }

    const v8f vzero = {0.f,0.f,0.f,0.f,0.f,0.f,0.f,0.f};
    v8f Hc[4], Gc[4];
    #pragma unroll
    for (int c = 0; c < 4; ++c) { Hc[c] = vzero; Gc[c] = vzero; }
    float rmax[8], rsum[8];
    #pragma unroll
    for (int v = 0; v < 8; ++v) { rmax[v] = -1e30f; rsum[v] = 0.f; }

    // =================== main loop over 64 column tiles ===================
    for (int mt = 0; mt < 64; ++mt) {
        const int col0 = mt * 16;
        float* xc = lds + (mt & 1) * 1088;           // current tile buffer

        // prefetch next tile into the other buffer (its readers finished at mt-1,
        // protected by the trailing barrier of that iteration)
        if (mt + 1 < 64) {
            float* xn = lds + ((mt + 1) & 1) * 1088;
            async_copy_b128(xn + tr * LDSTRIDE + tc,
                            Xs + (size_t)((mt + 1) * 16 + tr) * DD + tc);
            wait_async_le1();              // my portion of tile mt has landed
        } else {
            wait_async_le0();
        }
        __syncthreads();                   // everyone's portion of tile mt landed

        // S(16x16) = Xrows(16x64) @ Xc^T(64x16)   [fp32 WMMA, K=4 per op]
        v8f s = vzero;
        #pragma unroll
        for (int kk = 0; kk < 16; ++kk) {
            v2f b = *(const v2f*)(xc + ln16 * LDSTRIDE + 4 * kk + 2 * hi);
            s = __builtin_amdgcn_wmma_f32_16x16x4_f32(false, ax[kk], false, b,
                                                      (short)0, s, false, false);
        }

        // adjacency tile in C/D layout: [m=v+8hi][n=ln16]
        float av[8];
        #pragma unroll
        for (int v = 0; v < 8; ++v)
            av[v] = A[(size_t)(rowbase + v + 8*hi) * NN + col0 + ln16];

        // online softmax update (per-row butterflies stay within 16-lane halves)
        float pv[8], scale[8];
        #pragma unroll
        for (int v = 0; v < 8; ++v) {
            float tmax = s[v];
            tmax = fmaxf(tmax, __shfl_xor(tmax, 1, 32));
            tmax = fmaxf(tmax, __shfl_xor(tmax, 2, 32));
            tmax = fmaxf(tmax, __shfl_xor(tmax, 4, 32));
            tmax = fmaxf(tmax, __shfl_xor(tmax, 8, 32));
            float nm = fmaxf(rmax[v], tmax);
            scale[v] = __expf(rmax[v] - nm);
            rmax[v] = nm;
            float p = __expf(s[v] - nm);
            pv[v] = p;
            float ts = p;
            ts += __shfl_xor(ts, 1, 32);
            ts += __shfl_xor(ts, 2, 32);
            ts += __shfl_xor(ts, 4, 32);
            ts += __shfl_xor(ts, 8, 32);
            rsum[v] = rsum[v] * scale[v] + ts;
        }
        #pragma unroll
        for (int c = 0; c < 4; ++c)
            #pragma unroll
            for (int v = 0; v < 8; ++v) Hc[c][v] *= scale[v];

        // stage (A∘P) and A tiles: D-layout -> LDS (padded) for A-operand reads
        #pragma unroll
        for (int v = 0; v < 8; ++v) {
            wsc[      (v + 8*hi) * TSTRIDE + ln16] = pv[v] * av[v];
            wsc[320 + (v + 8*hi) * TSTRIDE + ln16] = av[v];
        }

        // H += (A∘P) @ Xc ;  G += A @ Xc     (K=16, N=64 in 4 chunks)
        #pragma unroll
        for (int c = 0; c < 4; ++c) {
            #pragma unroll
            for (int kk = 0; kk < 4; ++kk) {
                v2f apa = *(const v2f*)(wsc +       ln16 * TSTRIDE + 4 * kk + 2 * hi);
                v2f aa  = *(const v2f*)(wsc + 320 + ln16 * TSTRIDE + 4 * kk + 2 * hi);
                int k0 = 4 * kk + 2 * hi;
                v2f b;
                b.x = xc[(k0 + 0) * LDSTRIDE + 16 * c + ln16];
                b.y = xc[(k0 + 1) * LDSTRIDE + 16 * c + ln16];
                Hc[c] = __builtin_amdgcn_wmma_f32_16x16x4_f32(false, apa, false, b,
                                                              (short)0, Hc[c], false, false);
                Gc[c] = __builtin_amdgcn_wmma_f32_16x16x4_f32(false, aa, false, b,
                                                              (short)0, Gc[c], false, false);
            }
        }
        __syncthreads();   // all reads of tile mt done; next prefetch may overwrite
    }

    // =================== epilogue ===================
    // H = Hc / (sqrt(d)*rowsum) + 1e-8 * Gc   (St = softmax/8 + 1e-8, gated by A)
    float inv[8];
    #pragma unroll
    for (int v = 0; v < 8; ++v) inv[v] = 1.0f / (8.0f * rsum[v]);
    #pragma unroll
    for (int c = 0; c < 4; ++c)
        #pragma unroll
        for (int v = 0; v < 8; ++v) {
            float h = Hc[c][v] * inv[v] + 1e-8f * Gc[c][v];
            wsc[(v + 8*hi) * LDSTRIDE + 16 * c + ln16] = h;  // stage for A-operand reads
        }

    // O(16x64) = H(16x64) @ W^T(64x64)
    v8f O[4];
    #pragma unroll
    for (int c = 0; c < 4; ++c) O[c] = vzero;
    #pragma unroll
    for (int c = 0; c < 4; ++c) {
        #pragma unroll
        for (int kk = 0; kk < 16; ++kk) {
            v2f a = *(const v2f*)(wsc + ln16 * LDSTRIDE + 4 * kk + 2 * hi);
            v2f b = *(const v2f*)(wl + (16 * c + ln16) * LDSTRIDE + 4 * kk + 2 * hi);
            O[c] = __builtin_amdgcn_wmma_f32_16x16x4_f32(false, a, false, b,
                                                         (short)0, O[c], false, false);
        }
    }

    // ReLU + LayerNorm over D (64 values/row = 4 chunks x 16 lanes of one half)
    float mean[8], rstd[8];
    #pragma unroll
    for (int v = 0; v < 8; ++v) {
        float sm = 0.f, sq = 0.f;
        #pragma unroll
        for (int c = 0; c < 4; ++c) {
            float x = fmaxf(O[c][v], 0.0f);
            O[c][v] = x;
            sm += x; sq += x * x;
        }
        sm += __shfl_xor(sm, 1, 32);  sq += __shfl_xor(sq, 1, 32);
        sm += __shfl_xor(sm, 2, 32);  sq += __shfl_xor(sq, 2, 32);
        sm += __shfl_xor(sm, 4, 32);  sq += __shfl_xor(sq, 4, 32);
        sm += __shfl_xor(sm, 8, 32);  sq += __shfl_xor(sq, 8, 32);
        float mu  = sm * (1.0f / 64.0f);
        float var = sq * (1.0f / 64.0f) - mu * mu;
        mean[v] = mu;
        rstd[v] = rsqrtf(var + 1e-5f);
    }

    // normalize * gamma + beta, add residual X, store
    #pragma unroll
    for (int c = 0; c < 4; ++c)
        #pragma unroll
        for (int v = 0; v < 8; ++v) {
            int row = rowbase + v + 8 * hi;
            int col = 16 * c + ln16;
            float val = (O[c][v] - mean[v]) * rstd[v] * gmm[c] + bet[c];
            val += Xs[(size_t)row * DD + col];
            Os[(size_t)row * DD + col] = val;
        }
}

extern "C" void kernel_launch(void* const* d_in, const int* in_sizes, int n_in,
                              void* d_out, int out_size, void* d_ws, size_t ws_size,
                              hipStream_t stream) {
    (void)in_sizes; (void)n_in; (void)out_size; (void)d_ws; (void)ws_size;
    const float* X     = (const float*)d_in[0];
    const float* A     = (const float*)d_in[1];
    const float* W     = (const float*)d_in[2];
    const float* gamma = (const float*)d_in[3];
    const float* beta  = (const float*)d_in[4];
    float* out = (float*)d_out;

    dim3 grid(48 * 8);   // 48 (b,t) slices x 8 row-blocks of 128 rows
    dim3 block(256);     // 8 waves, 16 rows per wave
    hipLaunchKernelGGL(sdgcn_fused_kernel, grid, block, 0, stream,
                       X, A, W, gamma, beta, out);
}